// LanguageModelOldCriterion_57990648430739
// MI455X (gfx1250) — compile-verified
//
#include <hip/hip_runtime.h>
#include <hip/hip_bf16.h>

// LanguageModelOldCriterion: masked gather + global reduce.
//   input  : [B=256, L=18, MP1=9488] f32 log-probs
//   target : [B=256, S=17] int token ids (0 = pad; first zero counted, second breaks)
//   out    : scalar f32 loss = -(sum of masked gathers) / (num masked)
//
// Latency-bound design for MI455X (wave32): one thread per batch row, all 17
// gather loads issued independently/unconditionally with 32-bit element
// offsets (input < 2^31 bytes) so the backend emits SADDR+voffset
// global_load_b32 with scale_offset and the j*MP1 term folded into the 24-bit
// immediate; mask applied as a multiplier; wave32 shfl tree + LDS combine.

#define B_   256
#define L_   18
#define MP1_ 9488
#define T_   17   // = L-1 = S

__global__ __launch_bounds__(256, 1)
void lm_old_criterion_kernel(const float* __restrict__ inp,
                             const int*   __restrict__ tgt,
                             float*       __restrict__ out) {
    const unsigned int b = threadIdx.x;     // one thread per batch row; blockDim.x == B_

    // ---- load this row's 17 targets (contiguous; vectorizes to b128 loads) ----
    int tg[T_];
    const int* trow = tgt + b * (unsigned)T_;
    #pragma unroll
    for (int j = 0; j < T_; ++j) tg[j] = trow[j];

    // ---- issue all 17 gather loads independently, 32-bit element offsets ----
    // element offset of input[b, 1+j, tg[j]] = b*L*MP1 + MP1 + j*MP1 + tg[j]
    float vals[T_];
    const unsigned int rowBase = b * (unsigned)(L_ * MP1_) + (unsigned)MP1_;
    #pragma unroll
    for (int j = 0; j < T_; ++j) {
        const unsigned int eoff = rowBase + (unsigned)(j * MP1_) + (unsigned)tg[j];
        vals[j] = inp[eoff];               // zext(u32) -> saddr + voffset, scale_offset
    }

    // ---- mask: include position iff inclusive cumsum of zeros <= 1 ----
    float sum = 0.0f, cnt = 0.0f;
    int zc = 0;
    #pragma unroll
    for (int j = 0; j < T_; ++j) {
        zc += (tg[j] == 0) ? 1 : 0;
        const float m = (zc <= 1) ? 1.0f : 0.0f;
        sum = fmaf(m, vals[j], sum);
        cnt += m;
    }

    // ---- wave32 reduction (5 shfl steps) ----
    #pragma unroll
    for (int off = 16; off > 0; off >>= 1) {
        sum += __shfl_down(sum, off, 32);
        cnt += __shfl_down(cnt, off, 32);
    }

    // ---- cross-wave combine via LDS (8 waves) ----
    __shared__ float ssum[8];
    __shared__ float scnt[8];
    const int wave = threadIdx.x >> 5;
    const int lane = threadIdx.x & 31;
    if (lane == 0) { ssum[wave] = sum; scnt[wave] = cnt; }
    __syncthreads();

    if (threadIdx.x == 0) {
        float S = 0.0f, C = 0.0f;
        #pragma unroll
        for (int w = 0; w < 8; ++w) { S += ssum[w]; C += scnt[w]; }
        out[0] = -S / C;   // C >= 256 always (first position of each row is counted)
    }
}

extern "C" void kernel_launch(void* const* d_in, const int* in_sizes, int n_in,
                              void* d_out, int out_size, void* d_ws, size_t ws_size,
                              hipStream_t stream) {
    (void)in_sizes; (void)n_in; (void)out_size; (void)d_ws; (void)ws_size;
    const float* inp = (const float*)d_in[0];
    const int*   tgt = (const int*)d_in[1];
    float*       out = (float*)d_out;
    lm_old_criterion_kernel<<<1, 256, 0, stream>>>(inp, tgt, out);
}